// Matrix_Factorization_34376918237314
// MI455X (gfx1250) — compile-verified
//
#include <hip/hip_runtime.h>
#include <math.h>

// ---- problem constants (from reference) ----
#define DIM       256
#define BATCH     8192
#define NEGK      64
#define ALPHA_F   1.0f
#define DECAY_F   1e-4f

#define WAVES_PER_BLOCK 8
#define THREADS (WAVES_PER_BLOCK * 32)

typedef __attribute__((ext_vector_type(2))) float v2f;
typedef __attribute__((ext_vector_type(8))) float v8f;

// One wave32 per batch row.
//   scores[b, 0:64] = NegTile(64x256) * s_e(256)  via V_WMMA_F32_16X16X4_F32
//   (N columns of B replicated with s_e -> all 16 output columns identical; FLOP
//    waste is irrelevant, workload is an L2-resident gather problem)
__global__ __launch_bounds__(THREADS)
void mf_hardneg_kernel(const float* __restrict__ user_emb,
                       const float* __restrict__ item_emb,
                       const int*   __restrict__ users,
                       const int*   __restrict__ pos_items,
                       const int*   __restrict__ neg_items,
                       float* __restrict__ out_bpr,
                       float* __restrict__ out_ss_s,
                       float* __restrict__ out_ss_p,
                       float* __restrict__ out_ss_n)
{
    __shared__ float s_lds[WAVES_PER_BLOCK][DIM];

    const int lane = threadIdx.x & 31;
    const int wave = threadIdx.x >> 5;
    const int b    = blockIdx.x * WAVES_PER_BLOCK + wave;   // grid sized exactly: b < BATCH always

    // ---- stage user embedding row into LDS (8 f32 per lane) ----
    const float* __restrict__ srow = user_emb + (long)users[b] * DIM;
    {
        const float4* s4 = (const float4*)srow;
        float4 sa = s4[lane * 2 + 0];
        float4 sb = s4[lane * 2 + 1];
        float* sd = &s_lds[wave][lane * 8];
        ((float4*)sd)[0] = sa;
        ((float4*)sd)[1] = sb;
    }
    __syncthreads();

    const float* __restrict__ prow = item_emb + (long)pos_items[b] * DIM;
    __builtin_prefetch(prow + lane * 8, 0, 0);   // global_prefetch_b8: pull pos row toward L2

    // ---- hard-negative selection: 4 tiles of 16 negatives, K swept 4 at a time ----
    // A-matrix 16x4 f32 layout (ISA 7.12.2): lanes 0-15 hold M=0..15 with K={0,1} in
    // vgpr{0,1}; lanes 16-31 hold K={2,3}.  B 4x16: mirrored K mapping, N striped on lanes.
    const int  kk = (lane < 16) ? 0 : 2;     // K sub-offset for this lane half
    const int  m  = lane & 15;               // negative row within tile (A M-index)

    float best    = -INFINITY;
    int   bestIdx = 0;

    for (int t = 0; t < 4; ++t) {
        const int nid = neg_items[b * NEGK + t * 16 + m];
        const float* __restrict__ nrow = item_emb + (long)nid * DIM;

        v8f c = {0.f, 0.f, 0.f, 0.f, 0.f, 0.f, 0.f, 0.f};
        #pragma unroll 8
        for (int k0 = 0; k0 < DIM; k0 += 4) {
            v2f a;                                    // A: contiguous pair from negative row
            a.x = nrow[k0 + kk + 0];
            a.y = nrow[k0 + kk + 1];
            v2f bm;                                   // B: s_e chunk broadcast to all N columns
            bm.x = s_lds[wave][k0 + kk + 0];
            bm.y = s_lds[wave][k0 + kk + 1];
            c = __builtin_amdgcn_wmma_f32_16x16x4_f32(
                    /*neg_a=*/false, a, /*neg_b=*/false, bm,
                    /*c_mod=*/(short)0, c, /*reuse_a=*/false, /*reuse_b=*/false);
        }

        // C/D layout: lanes 0-15 hold M=0..7 in c[0..7], lanes 16-31 hold M=8..15.
        const int mbase = (lane < 16) ? 0 : 8;
        #pragma unroll
        for (int r = 0; r < 8; ++r) {
            float sc  = c[r];
            int   idx = t * 16 + mbase + r;
            if (sc > best || (sc == best && idx < bestIdx)) { best = sc; bestIdx = idx; }
        }
    }

    // butterfly argmax across the wave (tie-break: lowest index, matching jnp.argmax)
    #pragma unroll
    for (int off = 16; off > 0; off >>= 1) {
        float ob = __shfl_xor(best, off, 32);
        int   oi = __shfl_xor(bestIdx, off, 32);
        if (ob > best || (ob == best && oi < bestIdx)) { best = ob; bestIdx = oi; }
    }

    // ---- dimension-independent mixup + per-row partials (8 dims per lane, f32 VALU) ----
    const float* __restrict__ hrow = item_emb + (long)neg_items[b * NEGK + bestIdx] * DIM;

    float accPos = 0.f, accNeg = 0.f, ssS = 0.f, ssP = 0.f, ssN = 0.f;
    #pragma unroll
    for (int i = 0; i < 8; ++i) {
        const int d = lane * 8 + i;
        const float s = s_lds[wave][d];
        const float p = prow[d];
        const float h = hrow[d];
        const float ns = __expf(s * h);
        const float ps = ALPHA_F * __expf(s * p);
        const float w  = ns / (ns + ps);
        const float nm = (1.f - w) * p + w * h;    // synthesized negative
        accPos += s * p;
        accNeg += s * nm;
        ssS += s * s;
        ssP += p * p;
        ssN += nm * nm;
    }
    #pragma unroll
    for (int off = 16; off > 0; off >>= 1) {
        accPos += __shfl_xor(accPos, off, 32);
        accNeg += __shfl_xor(accNeg, off, 32);
        ssS    += __shfl_xor(ssS,    off, 32);
        ssP    += __shfl_xor(ssP,    off, 32);
        ssN    += __shfl_xor(ssN,    off, 32);
    }

    if (lane == 0) {
        const float x   = accPos - accNeg;
        const float sig = 1.f / (1.f + __expf(-x));
        out_bpr[b]  = logf(1e-5f + sig);
        out_ss_s[b] = ssS;
        out_ss_p[b] = ssP;
        out_ss_n[b] = ssN;
    }
}

// Deterministic single-block reduction (fixed tree order -> bitwise stable across replays).
__global__ __launch_bounds__(256)
void mf_reduce_kernel(const float* __restrict__ bpr,
                      const float* __restrict__ ss,
                      const float* __restrict__ sp,
                      const float* __restrict__ sn,
                      float* __restrict__ out)
{
    __shared__ float r0[256], r1[256], r2[256], r3[256];
    const int t = threadIdx.x;
    float a0 = 0.f, a1 = 0.f, a2 = 0.f, a3 = 0.f;
    for (int i = t; i < BATCH; i += 256) {
        a0 += bpr[i]; a1 += ss[i]; a2 += sp[i]; a3 += sn[i];
    }
    r0[t] = a0; r1[t] = a1; r2[t] = a2; r3[t] = a3;
    __syncthreads();
    for (int s = 128; s > 0; s >>= 1) {
        if (t < s) {
            r0[t] += r0[t + s]; r1[t] += r1[t + s];
            r2[t] += r2[t + s]; r3[t] += r3[t + s];
        }
        __syncthreads();
    }
    if (t == 0) {
        const float invB    = 1.f / (float)BATCH;
        const float bprLoss = -(r0[0] * invB);
        const float embLoss = (sqrtf(r1[0]) + sqrtf(r2[0]) + sqrtf(r3[0])) * invB;
        out[0] = bprLoss + embLoss * DECAY_F * invB;
    }
}

extern "C" void kernel_launch(void* const* d_in, const int* in_sizes, int n_in,
                              void* d_out, int out_size, void* d_ws, size_t ws_size,
                              hipStream_t stream)
{
    (void)in_sizes; (void)n_in; (void)out_size; (void)ws_size;

    const float* user_emb  = (const float*)d_in[0];
    const float* item_emb  = (const float*)d_in[1];
    const int*   users     = (const int*)  d_in[2];
    const int*   pos_items = (const int*)  d_in[3];
    const int*   neg_items = (const int*)  d_in[4];

    float* ws    = (float*)d_ws;          // 4 * BATCH floats = 128 KB of scratch
    float* w_bpr = ws + 0 * BATCH;
    float* w_ss  = ws + 1 * BATCH;
    float* w_sp  = ws + 2 * BATCH;
    float* w_sn  = ws + 3 * BATCH;

    mf_hardneg_kernel<<<BATCH / WAVES_PER_BLOCK, THREADS, 0, stream>>>(
        user_emb, item_emb, users, pos_items, neg_items, w_bpr, w_ss, w_sp, w_sn);

    mf_reduce_kernel<<<1, 256, 0, stream>>>(w_bpr, w_ss, w_sp, w_sn, (float*)d_out);
}